// PyramidROIAlign_1176821039144
// MI455X (gfx1250) — compile-verified
//
#include <hip/hip_runtime.h>
#include <hip/hip_bf16.h>

// PyramidROIAlign for MI455X (gfx1250, wave32).
//
// Roofline: ~100MB output stores + ~170MB pyramid reads (pyramid nearly fits
// the 192MB L2) => ~12us @ 23.3TB/s; FLOPs (~0.2G) are irrelevant. The kernel
// is organized around data movement, with the x-interpolation mapped onto the
// matrix pipe in full f32:
//  * 1 workgroup (256 thr = 8 waves) per box.
//  * Stage 1 (gather, VMEM b128): y-blend the <=14 needed feature columns
//    into LDS:  R[s][c] = (1-wy)*FM[y0,xc[s],c] + wy*FM[y1,xc[s],c]
//    NHWC -> channel-contiguous float4 loads (512B/wave/instruction).
//  * Stage 2 (matrix pipe): x-interp is a banded 7x14 x 14x256 f32 GEMM:
//    Out[j,c] = sum_s Kx[j,s]*R[s,c],  Kx[j,2j]=1-wx_j, Kx[j,2j+1]=wx_j.
//    4 chained V_WMMA_F32_16X16X4_F32 (K=16 >= 14), one 16-channel tile per
//    chain, 8 waves x 2 tiles = 256 channels.
//
// VGPR layout (ISA 7.12.2, 32-bit 16x4 A / 4x16 B, wave32):
//   A: vgpr0 = {lanes0-15: K=0, lanes16-31: K=2}, vgpr1 = {K=1, K=3}, lane%16 = M
//   B: vgpr0 = {lanes0-15: K=0, lanes16-31: K=2}, vgpr1 = {K=1, K=3}, lane%16 = N
//   C/D: vgpr v = row M=v (lanes0-15) / M=v+8 (lanes16-31), lane%16 = N

typedef float v2f __attribute__((ext_vector_type(2)));
typedef float v8f __attribute__((ext_vector_type(8)));

#define POOL    7
#define CCH     256
#define RROWS   16
#define RSTRIDE 272   // 256 + 16 floats pad: lane halves (s, s+2) hit disjoint
                      // LDS banks; row pitch 1088B stays 16B-aligned for b128

__global__ __launch_bounds__(256) void roi_align_wmma_kernel(
    const float* __restrict__ boxes,        // [B,N,4] (y1,x1,y2,x2) normalized
    const int*   __restrict__ image_shape,  // [2]
    const float* __restrict__ p2,           // [B,256,256,256]
    const float* __restrict__ p3,           // [B,128,128,256]
    const float* __restrict__ p4,           // [B, 64, 64,256]
    const float* __restrict__ p5,           // [B, 32, 32,256]
    float*       __restrict__ out,          // [B,N,7,7,256]
    int B, int N)
{
    __shared__ __align__(16) float Rbuf[RROWS * RSTRIDE]; // y-blended cols R[s][c]
    __shared__ float s_wx[POOL], s_wy[POOL];
    __shared__ int   s_x0[POOL], s_x1[POOL], s_y0[POOL], s_y1[POOL];
    __shared__ int   s_lvl;

    const int tid  = threadIdx.x;
    const int lane = tid & 31;
    const int bn   = blockIdx.x;
    const int b    = bn / N;

    // ---- per-box setup: FPN level + 7 sample coords (lanes 0..6 of wave 0) ----
    if (tid < POOL) {
        const float* bx = boxes + (size_t)bn * 4;
        const float y1 = bx[0], x1 = bx[1], y2 = bx[2], x2 = bx[3];
        const float ih = (float)image_shape[0], iw = (float)image_shape[1];
        float scaled = sqrtf(fmaxf((y2 - y1) * (x2 - x1), 1e-30f)) *
                       sqrtf(ih * iw) * (1.0f / 224.0f);
        int lvl = 4 + (int)rintf(log2f(scaled));
        lvl = max(2, min(5, lvl));
        const int S = 256 >> (lvl - 2);

        const float t  = (float)tid * (1.0f / 6.0f);
        const float ys = (y1 + (y2 - y1) * t) * (float)(S - 1);
        const float xs = (x1 + (x2 - x1) * t) * (float)(S - 1);
        const float y0f = floorf(ys), x0f = floorf(xs);
        s_wy[tid] = ys - y0f;
        s_wx[tid] = xs - x0f;
        int y0 = min(max((int)y0f, 0), S - 1);
        int x0 = min(max((int)x0f, 0), S - 1);
        s_y0[tid] = y0;  s_y1[tid] = min(y0 + 1, S - 1);
        s_x0[tid] = x0;  s_x1[tid] = min(x0 + 1, S - 1);
        if (tid == 0) s_lvl = lvl;
    }
    // zero the two unused K rows (s=14,15) so 0*Rbuf can't make NaNs
    for (int z = tid; z < 2 * RSTRIDE; z += 256)
        Rbuf[14 * RSTRIDE + z] = 0.0f;
    __syncthreads();

    const int lvl = s_lvl;
    const int S   = 256 >> (lvl - 2);
    const float* fm  = (lvl == 2) ? p2 : (lvl == 3) ? p3 : (lvl == 4) ? p4 : p5;
    const float* fmb = fm + (size_t)b * S * S * CCH;

    // ---- build banded A (Kx) fragments once per box ----
    const int mrow = lane & 15;          // M for A, N for B
    const int thi  = lane >> 4;          // 0: K slots {0,1}, 1: K slots {2,3}
    v2f afrag[4];
#pragma unroll
    for (int kk = 0; kk < 4; ++kk) {
        const int s0 = 4 * kk + (thi ? 2 : 0);
        float a0 = 0.0f, a1 = 0.0f;
        if (mrow < POOL) {
            const float w = s_wx[mrow];
            const int d0 = s0 - 2 * mrow;        // 0 -> (1-w), 1 -> w
            const int d1 = d0 + 1;
            a0 = (d0 == 0) ? (1.0f - w) : (d0 == 1) ? w : 0.0f;
            a1 = (d1 == 0) ? (1.0f - w) : (d1 == 1) ? w : 0.0f;
        }
        afrag[kk].x = a0;
        afrag[kk].y = a1;
    }

    const int wave = tid >> 5;

    // ---- per output row i: B128 gather+y-blend into LDS, then WMMA x-interp ----
    for (int i = 0; i < POOL; ++i) {
        const float wy = s_wy[i];
        const float* row0 = fmb + (size_t)s_y0[i] * S * CCH;
        const float* row1 = fmb + (size_t)s_y1[i] * S * CCH;

        // 14 columns x 64 float4 = 896 vector elements; 512B per wave-load
#pragma unroll
        for (int f = tid; f < 2 * POOL * (CCH / 4); f += 256) {
            const int s   = f >> 6;              // column slot 0..13
            const int c4  = (f & 63) << 2;       // channel base (float4)
            const int j   = s >> 1;
            const int col = (s & 1) ? s_x1[j] : s_x0[j];
            const float4 v0 = *(const float4*)(row0 + (size_t)col * CCH + c4);
            const float4 v1 = *(const float4*)(row1 + (size_t)col * CCH + c4);
            float4 r;
            r.x = v0.x + (v1.x - v0.x) * wy;
            r.y = v0.y + (v1.y - v0.y) * wy;
            r.z = v0.z + (v1.z - v0.z) * wy;
            r.w = v0.w + (v1.w - v0.w) * wy;
            *(float4*)(&Rbuf[s * RSTRIDE + c4]) = r;
        }
        __syncthreads();

#pragma unroll
        for (int t = 0; t < 2; ++t) {
            const int cbase = (wave * 2 + t) * 16;
            v8f acc = {0.f, 0.f, 0.f, 0.f, 0.f, 0.f, 0.f, 0.f};
#if defined(__gfx1250__) && __has_builtin(__builtin_amdgcn_wmma_f32_16x16x4_f32)
#pragma unroll
            for (int kk = 0; kk < 4; ++kk) {
                const int s0 = 4 * kk + (thi ? 2 : 0);
                v2f bfrag;
                bfrag.x = Rbuf[(s0    ) * RSTRIDE + cbase + mrow];
                bfrag.y = Rbuf[(s0 + 1) * RSTRIDE + cbase + mrow];
                acc = __builtin_amdgcn_wmma_f32_16x16x4_f32(
                    false, afrag[kk], false, bfrag, (short)0, acc, false, false);
            }
            if (lane < 16) {
                const size_t obase =
                    (((size_t)bn * POOL + i) * POOL) * CCH + cbase + lane;
#pragma unroll
                for (int j = 0; j < POOL; ++j)
                    out[obase + (size_t)j * CCH] = acc[j];
            }
#else
            // VALU fallback (also taken by the host pass): direct x-blend
            if (t == 0) {
                for (int f = tid; f < POOL * CCH; f += 256) {
                    const int j = f / CCH, c = f - j * CCH;
                    const float r0 = Rbuf[(2 * j    ) * RSTRIDE + c];
                    const float r1 = Rbuf[(2 * j + 1) * RSTRIDE + c];
                    out[(((size_t)bn * POOL + i) * POOL + j) * CCH + c] =
                        r0 + (r1 - r0) * s_wx[j];
                }
            }
            (void)acc; (void)cbase;
#endif
        }
        __syncthreads();   // Rbuf reused next i
    }
}

extern "C" void kernel_launch(void* const* d_in, const int* in_sizes, int n_in,
                              void* d_out, int out_size, void* d_ws, size_t ws_size,
                              hipStream_t stream) {
    const float* boxes       = (const float*)d_in[0];
    const int*   image_shape = (const int*)d_in[1];
    const float* p2          = (const float*)d_in[2];
    const float* p3          = (const float*)d_in[3];
    const float* p4          = (const float*)d_in[4];
    const float* p5          = (const float*)d_in[5];
    float*       out         = (float*)d_out;

    const int BN = in_sizes[0] / 4;                       // B*N
    int B = in_sizes[2] / (256 * 256 * 256);              // p2 = [B,256,256,256]
    if (B <= 0) B = 1;
    const int N = BN / B;

    roi_align_wmma_kernel<<<BN, 256, 0, stream>>>(
        boxes, image_shape, p2, p3, p4, p5, out, B, N);
}